// TopK_custom_38826504355899
// MI455X (gfx1250) — compile-verified
//
#include <hip/hip_runtime.h>
#include <math.h>

// Problem constants fixed by the reference setup: scores (4096, 2048) f32,
// K=64, EPS=0.1, MAX_ITER=50, anchors {0,1}.
#define NCOLS 2048
#define KSEL 64.0f
#define NMK 1984.0f          // n - K
#define ITERS 50
#define ROWS_PER_BLOCK 8     // 8 waves (wave32) per 256-thread block, 1 row/wave
#define RED_BLOCKS 512

typedef __attribute__((ext_vector_type(4))) unsigned int v4u;
typedef __attribute__((ext_vector_type(8))) int v8i;
typedef __attribute__((ext_vector_type(4))) int v4i;

__device__ __forceinline__ float wave_sum32(float x) {
#pragma unroll
  for (int off = 16; off > 0; off >>= 1) x += __shfl_xor(x, off, 32);
  return x;
}

// ---------------- Pass 1a: per-block max / min(ignoring -inf) / any(-inf) ----
__global__ __launch_bounds__(256) void minmax_partial(const float* __restrict__ x,
                                                      long long total,
                                                      float* __restrict__ part) {
  const long long n4 = total >> 2;
  const float NINF = -__builtin_inff();
  const float PINF = __builtin_inff();
  float mx = NINF, mn = PINF, fi = 0.0f;
  const float4* x4 = (const float4*)x;
  const long long stride = (long long)gridDim.x * blockDim.x;
  for (long long i = (long long)blockIdx.x * blockDim.x + threadIdx.x; i < n4; i += stride) {
    const float4 v = x4[i];
    const float vv[4] = {v.x, v.y, v.z, v.w};
#pragma unroll
    for (int j = 0; j < 4; ++j) {
      const float s = vv[j];
      mx = fmaxf(mx, s);
      const bool ni = (s == NINF);
      fi = ni ? 1.0f : fi;
      mn = fminf(mn, ni ? PINF : s);
    }
  }
  if (blockIdx.x == 0 && threadIdx.x == 0) {   // scalar tail (none for 8M, kept for safety)
    for (long long i = n4 << 2; i < total; ++i) {
      const float s = x[i];
      mx = fmaxf(mx, s);
      const bool ni = (s == NINF);
      fi = ni ? 1.0f : fi;
      mn = fminf(mn, ni ? PINF : s);
    }
  }
  __shared__ float smx[256], smn[256], sfi[256];
  const int t = threadIdx.x;
  smx[t] = mx; smn[t] = mn; sfi[t] = fi;
  __syncthreads();
  for (int s2 = 128; s2 > 0; s2 >>= 1) {
    if (t < s2) {
      smx[t] = fmaxf(smx[t], smx[t + s2]);
      smn[t] = fminf(smn[t], smn[t + s2]);
      sfi[t] = fmaxf(sfi[t], sfi[t + s2]);
    }
    __syncthreads();
  }
  if (t == 0) {
    part[3 * blockIdx.x + 0] = smx[0];
    part[3 * blockIdx.x + 1] = smn[0];
    part[3 * blockIdx.x + 2] = sfi[0];
  }
}

// ---------------- Pass 1b: finalize scalars: alpha = 1/(EPS*maxC), filled ----
__global__ __launch_bounds__(256) void minmax_final(const float* __restrict__ part,
                                                    int nPart,
                                                    float* __restrict__ scal) {
  const float NINF = -__builtin_inff();
  const float PINF = __builtin_inff();
  float mx = NINF, mn = PINF, fi = 0.0f;
  for (int i = threadIdx.x; i < nPart; i += 256) {
    mx = fmaxf(mx, part[3 * i + 0]);
    mn = fminf(mn, part[3 * i + 1]);
    fi = fmaxf(fi, part[3 * i + 2]);
  }
  __shared__ float smx[256], smn[256], sfi[256];
  const int t = threadIdx.x;
  smx[t] = mx; smn[t] = mn; sfi[t] = fi;
  __syncthreads();
  for (int s2 = 128; s2 > 0; s2 >>= 1) {
    if (t < s2) {
      smx[t] = fmaxf(smx[t], smx[t + s2]);
      smn[t] = fminf(smn[t], smn[t + s2]);
      sfi[t] = fmaxf(sfi[t], sfi[t + s2]);
    }
    __syncthreads();
  }
  if (t == 0) {
    const float maxS = smx[0], minS = smn[0];
    const float filled = minS - (maxS - minS);
    // max(C) is attained at the extreme s values ((s-a)^2 convex, a in {0,1})
    const float emin = (sfi[0] > 0.0f) ? filled : minS;  // filled <= minS
    const float emax = maxS;
    const float c1 = fmaxf(emax * emax, (emax - 1.0f) * (emax - 1.0f));
    const float c2 = fmaxf(emin * emin, (emin - 1.0f) * (emin - 1.0f));
    const float maxC = fmaxf(c1, c2);
    scal[0] = 10.0f / maxC;   // alpha = 1/(EPS*maxC), EPS = 0.1
    scal[1] = filled;
  }
}

// ---------------- Pass 2: Sinkhorn, one wave32 per row ----------------------
__global__ __launch_bounds__(256) void sinkhorn_kernel(const float* __restrict__ scores,
                                                       float* __restrict__ out,
                                                       const float* __restrict__ scal,
                                                       int bs) {
  __shared__ float sh[ROWS_PER_BLOCK * NCOLS];   // 64 KB: 8 rows staged by the TDM
  const int wave = threadIdx.x >> 5;
  const int lane = threadIdx.x & 31;
  const int row0 = blockIdx.x * ROWS_PER_BLOCK;
  const int row  = row0 + wave;

  const float alpha  = scal[0];
  const float filled = scal[1];

  // ---- Tensor Data Mover: DMA the 2048x8 f32 tile into LDS (one issue/block)
  if (wave == 0) {
    const unsigned long long ga =
        (unsigned long long)(const void*)(scores + (long long)row0 * NCOLS);
    v4u g0;
    g0.x = 1u;                                    // count=1 descriptor, load mode
    g0.y = 0u;                                    // lds_addr = 0 (sh is first/only LDS alloc)
    g0.z = (unsigned)ga;                          // global_addr[31:0]
    g0.w = ((unsigned)(ga >> 32) & 0x01FFFFFFu)   // global_addr[56:32]
           | (2u << 30);                          // type = 2 ("image")
    v8i g1;
    g1.s0 = 0x00020000;                           // workgroup_mask=0, data_size=2 (4B)
    g1.s1 = (int)(0x0800u << 16);                 // tensor_dim0[15:0] = 2048
    g1.s2 = (int)(8u << 16);                      // tensor_dim0[31:16]=0, tensor_dim1=8
    g1.s3 = (int)(0x0800u << 16);                 // tensor_dim1 hi=0, tile_dim0 = 2048
    g1.s4 = 8;                                    // tile_dim1 = 8, tile_dim2 = 0
    g1.s5 = 2048;                                 // tensor_dim0_stride = 2048
    g1.s6 = 0;                                    // stride hi / tensor_dim1_stride lo
    g1.s7 = 0;
    v4i gz4 = {0, 0, 0, 0};
    v8i gz8 = {0, 0, 0, 0, 0, 0, 0, 0};
    // 6-arg form (clang-23 / therock-10.0 headers): (g0, g1, g2, g3, g4, cpol)
    __builtin_amdgcn_tensor_load_to_lds(g0, g1, gz4, gz4, gz8, 0);
    __builtin_amdgcn_s_wait_tensorcnt(0);
  }
  __syncthreads();

  // ---- Build r_i = exp((2 s_i - 1) * alpha); 64 elements/lane in VGPRs ----
  float r[64];
  const int base = wave * NCOLS + lane * 4;
#pragma unroll
  for (int c = 0; c < 16; ++c) {
    const float4 sv = *(const float4*)&sh[base + c * 128];
    const float ss[4] = {sv.x, sv.y, sv.z, sv.w};
#pragma unroll
    for (int j = 0; j < 4; ++j) {
      float s = ss[j];
      if (s == -__builtin_inff()) s = filled;
      const float t = fmaf(2.0f * alpha, s, -alpha);   // (2s-1)*alpha, |t| <= ~20
      r[c * 4 + j] = __expf(t);                        // v_exp_f32 path
    }
  }

  // ---- 50 Sinkhorn iterations; v_rcp_f32-throughput-bound inner loop ------
  // 4-way split accumulators keep the add/fma dependence chains at length 16
  // so the TRANS pipe (rcp) and VALU co-execute without latency stalls.
  float v0 = 0.5f, v1 = 0.5f, p0 = 0.5f, p1 = 0.5f;
  for (int it = 0; it < ITERS; ++it) {
    float a0x = 0.0f, a0y = 0.0f, a0z = 0.0f, a0w = 0.0f;
    float a1x = 0.0f, a1y = 0.0f, a1z = 0.0f, a1w = 0.0f;
#pragma unroll
    for (int e = 0; e < 64; e += 4) {
      const float w0 = __builtin_amdgcn_rcpf(fmaf(r[e + 0], v1, v0));
      const float w1 = __builtin_amdgcn_rcpf(fmaf(r[e + 1], v1, v0));
      const float w2 = __builtin_amdgcn_rcpf(fmaf(r[e + 2], v1, v0));
      const float w3 = __builtin_amdgcn_rcpf(fmaf(r[e + 3], v1, v0));
      a0x += w0; a0y += w1; a0z += w2; a0w += w3;
      a1x = fmaf(r[e + 0], w0, a1x);
      a1y = fmaf(r[e + 1], w1, a1y);
      a1z = fmaf(r[e + 2], w2, a1z);
      a1w = fmaf(r[e + 3], w3, a1w);
    }
    float a0 = (a0x + a0y) + (a0z + a0w);
    float a1 = (a1x + a1y) + (a1z + a1w);
    a0 = wave_sum32(a0);
    a1 = wave_sum32(a1);
    p0 = v0; p1 = v1;                                  // keep v^(49) for output
    v0 = KSEL / a0;                                    // v0' = K / T0
    v1 = NMK / a1;                                     // v1' = (n-K) / T1
  }

  // ---- A_i = v0^(50) / (v0^(49) + r_i * v1^(49)) ----
  if (row < bs) {
    float4* o4 = (float4*)(out + (long long)row * NCOLS);
#pragma unroll
    for (int c = 0; c < 16; ++c) {
      float4 o;
      o.x = v0 * __builtin_amdgcn_rcpf(fmaf(r[c * 4 + 0], p1, p0));
      o.y = v0 * __builtin_amdgcn_rcpf(fmaf(r[c * 4 + 1], p1, p0));
      o.z = v0 * __builtin_amdgcn_rcpf(fmaf(r[c * 4 + 2], p1, p0));
      o.w = v0 * __builtin_amdgcn_rcpf(fmaf(r[c * 4 + 3], p1, p0));
      o4[c * 32 + lane] = o;
    }
  }
}

extern "C" void kernel_launch(void* const* d_in, const int* in_sizes, int n_in,
                              void* d_out, int out_size, void* d_ws, size_t ws_size,
                              hipStream_t stream) {
  const float* scores = (const float*)d_in[0];
  float* out = (float*)d_out;
  float* ws = (float*)d_ws;

  const long long total = (long long)in_sizes[0];
  const int bs = (int)(total / NCOLS);               // 4096 for the reference setup

  float* part = ws;                                   // RED_BLOCKS * 3 floats
  float* scal = ws + 2048;                            // [alpha, filled]

  minmax_partial<<<RED_BLOCKS, 256, 0, stream>>>(scores, total, part);
  minmax_final<<<1, 256, 0, stream>>>(part, RED_BLOCKS, scal);
  sinkhorn_kernel<<<bs / ROWS_PER_BLOCK, 256, 0, stream>>>(scores, out, scal, bs);
}